// RoIAwarePool3d_19550691131702
// MI455X (gfx1250) — compile-verified
//
#include <hip/hip_runtime.h>

// RoIAwarePool3d (max mode) for MI455X / gfx1250.
//
// d_in[0] = rois        (128*7 f32)  [cx,cy,z_bottom,dx,dy,dz,ry]
// d_in[1] = pts         (160000*3 f32)
// d_in[2] = pts_feature (160000*64 f32)
// d_out   = (128,12,12,12,64) f32 = 14,155,776 elements
//
// Pipeline: init(0) -> scatter (TDM-staged ROI table in LDS, bounding-circle
// early-out, global_atomic_max_u32 on order-preserving float encodings) ->
// finalize (decode, sentinel 0 -> 0.0f). Output (56.6 MB) fits in the 192 MB
// L2, so the init/atomic/finalize chain stays on-chip; HBM sees ~60 MB.

#define RP_OUT_X 12
#define RP_OUT_Y 12
#define RP_OUT_Z 12
#define RP_NVOX  (RP_OUT_X * RP_OUT_Y * RP_OUT_Z)   // 1728
#define RP_NROIS 128
#define RP_NPTS  160000
#define RP_CHN   64

typedef unsigned int u32x4 __attribute__((ext_vector_type(4)));
typedef int          i32x4 __attribute__((ext_vector_type(4)));
typedef int          i32x8 __attribute__((ext_vector_type(8)));

// Order-preserving float -> u32 map: monotone, so u32 max == float max.
__device__ __forceinline__ unsigned rp_enc(float f) {
    unsigned b = __float_as_uint(f);
    return (b & 0x80000000u) ? ~b : (b | 0x80000000u);
}
__device__ __forceinline__ float rp_dec(unsigned v) {
    if (v == 0u) return 0.0f;  // sentinel: no point ever landed here
    return (v & 0x80000000u) ? __uint_as_float(v & 0x7FFFFFFFu)
                             : __uint_as_float(~v);
}

__global__ void rp3d_init(unsigned* __restrict__ o, int n4) {
    int i = blockIdx.x * blockDim.x + threadIdx.x;
    if (i < n4) reinterpret_cast<uint4*>(o)[i] = make_uint4(0u, 0u, 0u, 0u);
}

__global__ void rp3d_finalize(unsigned* __restrict__ o, int n4) {
    int i = blockIdx.x * blockDim.x + threadIdx.x;
    if (i < n4) {
        uint4 v = reinterpret_cast<uint4*>(o)[i];
        float4 f;
        f.x = rp_dec(v.x); f.y = rp_dec(v.y); f.z = rp_dec(v.z); f.w = rp_dec(v.w);
        reinterpret_cast<float4*>(o)[i] = f;
    }
}

__global__ void __launch_bounds__(256)
rp3d_scatter(const float* __restrict__ rois,
             const float* __restrict__ pts,
             const float* __restrict__ feat,
             unsigned* __restrict__ out_u) {
    __shared__ float  s_raw[RP_NROIS * 7];   // raw ROI table, filled by TDM DMA
    __shared__ float4 s_cc[RP_NROIS];        // hot: cx, cy, cz, rmax^2
    __shared__ float  s_full[RP_NROIS * 8];  // cold: hdx,hdy,dz,ca,sa,ivx,ivy,ivz

#if __has_builtin(__builtin_amdgcn_tensor_load_to_lds)
    if (threadIdx.x < 32) {
        // Tensor DMA descriptor (ISA ch.8): 1-D tensor of 896 f32 elements,
        // data_size=4B (code 2), tile==tensor, descriptor type=2 ("image").
        unsigned long long ga = (unsigned long long)rois;
        unsigned lds_addr = (unsigned)(unsigned long long)(void*)s_raw; // low 32b of flat addr = LDS byte offset
        const unsigned NF = RP_NROIS * 7;   // 896 elements
        u32x4 g0;
        g0[0] = 1u;                                             // count=1 (valid user D#)
        g0[1] = lds_addr;                                       // lds_addr [63:32]
        g0[2] = (unsigned)ga;                                   // global_addr [95:64]
        g0[3] = (unsigned)((ga >> 32) & 0x1FFFFFFull)           // global_addr [120:96]
              | (2u << 30);                                     // type=2
        i32x8 g1;
        g1[0] = (int)(2u << 16);        // data_size=2 (4 bytes); wg_mask=0 (not in cluster)
        g1[1] = (int)(NF << 16);        // tensor_dim0[15:0]
        g1[2] = (int)(1u << 16);        // tensor_dim0[31:16]=0, tensor_dim1=1
        g1[3] = (int)(NF << 16);        // tile_dim0 = 896
        g1[4] = 1;                      // tile_dim1=1, tile_dim2=0
        g1[5] = (int)NF;                // tensor_dim0_stride low 32
        g1[6] = (int)(NF << 16);        // stride0 hi=0, tensor_dim1_stride lo16
        g1[7] = 0;
        i32x4 gz = {0, 0, 0, 0};
#if defined(__clang_major__) && __clang_major__ >= 23
        i32x8 gz8 = {0, 0, 0, 0, 0, 0, 0, 0};
        __builtin_amdgcn_tensor_load_to_lds(g0, g1, gz, gz, gz8, 0);
#else
        __builtin_amdgcn_tensor_load_to_lds(g0, g1, gz, gz, 0);
#endif
        __builtin_amdgcn_s_wait_tensorcnt(0);
    }
#else
    for (int i = threadIdx.x; i < RP_NROIS * 7; i += blockDim.x) s_raw[i] = rois[i];
#if __has_builtin(__builtin_amdgcn_s_wait_tensorcnt)
    __builtin_amdgcn_s_wait_tensorcnt(0);
#endif
#endif
    __syncthreads();

    // Derive per-ROI constants once per block (sincos 128x/block, not per point).
    if (threadIdx.x < RP_NROIS) {
        int r = threadIdx.x;
        float cx = s_raw[r * 7 + 0], cy = s_raw[r * 7 + 1], cz = s_raw[r * 7 + 2];
        float dx = s_raw[r * 7 + 3], dy = s_raw[r * 7 + 4], dz = s_raw[r * 7 + 5];
        float ry = s_raw[r * 7 + 6];
        float hdx = 0.5f * dx, hdy = 0.5f * dy;
        // Rotation preserves the 2-D norm: inside-box => sx^2+sy^2 < hdx^2+hdy^2.
        s_cc[r] = make_float4(cx, cy, cz, fmaf(hdx, hdx, hdy * hdy));
        float* F = s_full + r * 8;
        F[0] = hdx;
        F[1] = hdy;
        F[2] = dz;
        F[3] = cosf(ry);    // cos(-ry) == cos(ry)
        F[4] = -sinf(ry);   // sin(-ry)
        F[5] = (float)RP_OUT_X / dx;
        F[6] = (float)RP_OUT_Y / dy;
        F[7] = (float)RP_OUT_Z / dz;
    }
    __syncthreads();

    int tid    = blockIdx.x * blockDim.x + threadIdx.x;
    int stride = gridDim.x * blockDim.x;
    for (int p = tid; p < RP_NPTS; p += stride) {
        const float* pp = pts + 3ull * (unsigned)p;   // -> global_load_b96
        float px = pp[0], py = pp[1], pz = pp[2];

#pragma unroll 4
        for (int r = 0; r < RP_NROIS; ++r) {
            float4 cc = s_cc[r];                      // one ds_load_b128 (fast path)
            float sx = px - cc.x;
            float sy = py - cc.y;
            float sz = pz - cc.z;
            float d2 = fmaf(sx, sx, sy * sy);
            // Conservative bounding-circle + z-bottom early-out (~0.2% pass rate).
            if (d2 < cc.w && sz >= 0.0f) {
                const float* F = s_full + r * 8;
                float hdx = F[0], hdy = F[1], dz = F[2];
                float ca = F[3], sa = F[4];
                float lx = fmaf(sx, ca, -sy * sa);    // rotate into ROI-local frame
                float ly = fmaf(sx, sa,  sy * ca);
                if (sz <= dz && fabsf(lx) < hdx && fabsf(ly) < hdy) {
                    int xi = (int)floorf((lx + hdx) * F[5]);
                    int yi = (int)floorf((ly + hdy) * F[6]);
                    int zi = (int)floorf(sz * F[7]);
                    xi = min(RP_OUT_X - 1, max(0, xi));
                    yi = min(RP_OUT_Y - 1, max(0, yi));
                    zi = min(RP_OUT_Z - 1, max(0, zi));
                    int seg = (xi * RP_OUT_Y + yi) * RP_OUT_Z + zi;
                    unsigned long long base =
                        ((unsigned long long)r * RP_NVOX + (unsigned)seg) * RP_CHN;
                    const float4* pf =
                        reinterpret_cast<const float4*>(feat + (unsigned long long)p * RP_CHN);
                    unsigned* po = out_u + base;
#pragma unroll
                    for (int c4 = 0; c4 < RP_CHN / 4; ++c4) {
                        float4 f = pf[c4];                       // global_load_b128
                        atomicMax(po + 4 * c4 + 0, rp_enc(f.x)); // global_atomic_max_u32
                        atomicMax(po + 4 * c4 + 1, rp_enc(f.y));
                        atomicMax(po + 4 * c4 + 2, rp_enc(f.z));
                        atomicMax(po + 4 * c4 + 3, rp_enc(f.w));
                    }
                }
            }
        }
    }
}

extern "C" void kernel_launch(void* const* d_in, const int* in_sizes, int n_in,
                              void* d_out, int out_size, void* d_ws, size_t ws_size,
                              hipStream_t stream) {
    (void)in_sizes; (void)n_in; (void)d_ws; (void)ws_size;
    const float* rois = (const float*)d_in[0];
    const float* pts  = (const float*)d_in[1];
    const float* feat = (const float*)d_in[2];
    unsigned* out_u   = (unsigned*)d_out;

    int n4 = out_size / 4;  // out_size = 14,155,776, divisible by 4
    dim3 blk(256);

    rp3d_init<<<dim3((n4 + 255) / 256), blk, 0, stream>>>(out_u, n4);
    rp3d_scatter<<<dim3((RP_NPTS + 255) / 256), blk, 0, stream>>>(rois, pts, feat, out_u);
    rp3d_finalize<<<dim3((n4 + 255) / 256), blk, 0, stream>>>(out_u, n4);
}